// MultiBoxLoss_50869592654785
// MI455X (gfx1250) — compile-verified
//
#include <hip/hip_runtime.h>
#include <stdint.h>

#define B_      64
#define P_      8732
#define C_      81
#define O_      16
#define ROWS_B  128                            /* 8 waves x 16 rows */
#define TILES_B ((P_ + ROWS_B - 1) / ROWS_B)   /* 69 */
#define SORT_N  16384

typedef __attribute__((ext_vector_type(2))) float v2f;
typedef __attribute__((ext_vector_type(8))) float v8f;

// ---------------------------------------------------------------------------
// Kernel A: per-batch matching + localization (smooth-L1) loss partials
// One block per batch image. bto/bti kept fully in LDS (CDNA5: 320KB/WGP).
// ---------------------------------------------------------------------------
__global__ __launch_bounds__(256)
void match_kernel(const float* __restrict__ loc_data,
                  const float* __restrict__ priors,
                  const float* __restrict__ targets,
                  int*   __restrict__ conf_t,      // [B*P]
                  int*   __restrict__ num_pos,     // [B]
                  float* __restrict__ lossl_part)  // [B]
{
    extern __shared__ char smem[];
    float* s_bto  = (float*)smem;                                  // P_
    int*   s_bti  = (int*)  (s_bto + P_);                          // P_
    float* s_tr   = (float*)(s_bti + P_);                          // 16*4
    float* s_area = s_tr + 64;                                     // 16
    int*   s_lab  = (int*)  (s_area + 16);                         // 16
    unsigned long long* s_best = (unsigned long long*)(s_lab + 16);// 16 (8B aligned)
    float* s_redf = (float*)(s_best + 16);                         // 256
    int*   s_redi = (int*)  (s_redf + 256);                        // 256

    const int b   = blockIdx.x;
    const int tid = threadIdx.x;

    if (tid < O_) {
        const float* t = targets + (size_t)(b * O_ + tid) * 8;
        float x0 = t[3], y0 = t[4], x1 = t[5], y1 = t[6];
        s_tr[tid * 4 + 0] = x0; s_tr[tid * 4 + 1] = y0;
        s_tr[tid * 4 + 2] = x1; s_tr[tid * 4 + 3] = y1;
        s_area[tid] = (x1 - x0) * (y1 - y0);
        s_lab[tid]  = (int)t[1];
        s_best[tid] = 0ull;
    }
    __syncthreads();

    // Pass 1: IoU; per-prior best truth (first-max ties) + local per-truth
    // best prior, merged via packed u64 LDS atomicMax (ds_max_u64).
    float        lbo[O_];
    unsigned int lbp[O_];
#pragma unroll
    for (int o = 0; o < O_; ++o) { lbo[o] = -1.0f; lbp[o] = 0u; }

    for (int p = tid; p < P_; p += 256) {
        float q0 = priors[p * 4 + 0], q1 = priors[p * 4 + 1];
        float q2 = priors[p * 4 + 2], q3 = priors[p * 4 + 3];
        // replicate reference: point_form(center_size(priors)) round-trip
        float pcx = (q0 + q2) * 0.5f, pcy = (q1 + q3) * 0.5f;
        float pw  = q2 - q0,          ph  = q3 - q1;
        float px0 = pcx - pw * 0.5f,  py0 = pcy - ph * 0.5f;
        float px1 = pcx + pw * 0.5f,  py1 = pcy + ph * 0.5f;
        float areap = (px1 - px0) * (py1 - py0);

        float bov = -1.0f; int bix = 0;
#pragma unroll
        for (int o = 0; o < O_; ++o) {
            float ix0 = fmaxf(s_tr[o * 4 + 0], px0);
            float iy0 = fmaxf(s_tr[o * 4 + 1], py0);
            float ix1 = fminf(s_tr[o * 4 + 2], px1);
            float iy1 = fminf(s_tr[o * 4 + 3], py1);
            float iw = fmaxf(ix1 - ix0, 0.0f);
            float ih = fmaxf(iy1 - iy0, 0.0f);
            float inter = iw * ih;
            float iou = inter / (s_area[o] + areap - inter);
            if (iou > bov) { bov = iou; bix = o; }        // first-max tie-break
            if (iou > lbo[o]) { lbo[o] = iou; lbp[o] = (unsigned)p; }
        }
        s_bto[p] = bov;
        s_bti[p] = bix;
    }
#pragma unroll
    for (int o = 0; o < O_; ++o) {
        unsigned long long key =
            ((unsigned long long)__float_as_uint(lbo[o]) << 32) |
            (unsigned long long)(0xFFFFFFFFu - lbp[o]);
        atomicMax(&s_best[o], key);
    }
    __syncthreads();

    // Override scatter, sequential (last-wins) to match scatter semantics
    if (tid == 0) {
        for (int o = 0; o < O_; ++o) {
            unsigned long long key = s_best[o];
            int   bp = (int)(0xFFFFFFFFu - (unsigned)(key & 0xFFFFFFFFull));
            float ov = __uint_as_float((unsigned)(key >> 32));
            s_bti[bp] = o;
            s_bto[bp] = ov;
        }
    }
    __syncthreads();

    // Pass 2: conf_t, encode, smooth-L1
    int np = 0; float ll = 0.0f;
    for (int p = tid; p < P_; p += 256) {
        float q0 = priors[p * 4 + 0], q1 = priors[p * 4 + 1];
        float q2 = priors[p * 4 + 2], q3 = priors[p * 4 + 3];
        float pcx = (q0 + q2) * 0.5f, pcy = (q1 + q3) * 0.5f;
        float pw  = q2 - q0,          ph  = q3 - q1;

        int   ti  = s_bti[p];
        float bto = s_bto[p];
        int conf = (bto < 0.5f) ? 0 : s_lab[ti];
        conf_t[(size_t)b * P_ + p] = conf;
        if (conf > 0) {
            ++np;
            float m0 = s_tr[ti * 4 + 0], m1 = s_tr[ti * 4 + 1];
            float m2 = s_tr[ti * 4 + 2], m3 = s_tr[ti * 4 + 3];
            float g0 = ((m0 + m2) * 0.5f - pcx) / (0.1f * pw);
            float g1 = ((m1 + m3) * 0.5f - pcy) / (0.1f * ph);
            float g2 = logf((m2 - m0) / pw) / 0.2f;
            float g3 = logf((m3 - m1) / ph) / 0.2f;
            const float* L = loc_data + ((size_t)b * P_ + p) * 4;
            float g[4] = {g0, g1, g2, g3};
#pragma unroll
            for (int j = 0; j < 4; ++j) {
                float d = L[j] - g[j];
                float a = fabsf(d);
                ll += (a < 1.0f) ? 0.5f * d * d : a - 0.5f;
            }
        }
    }
    s_redf[tid] = ll; s_redi[tid] = np;
    __syncthreads();
    for (int s = 128; s > 0; s >>= 1) {
        if (tid < s) { s_redf[tid] += s_redf[tid + s]; s_redi[tid] += s_redi[tid + s]; }
        __syncthreads();
    }
    if (tid == 0) { num_pos[b] = s_redi[0]; lossl_part[b] = s_redf[0]; }
}

// ---------------------------------------------------------------------------
// Kernel B: cross-entropy per prior.
//  - stage 128 conf rows (~41KB) into LDS via CDNA5 async global->LDS loads
//  - per wave: 16 rows; Sum(exp) via chained V_WMMA_F32_16X16X4_F32 with
//    B = ones (each WMMA reduces a K=4 slice of all 16 rows at once)
// ---------------------------------------------------------------------------
__global__ __launch_bounds__(256)
void ce_kernel(const float* __restrict__ conf_data,
               const int*   __restrict__ conf_t,
               float* __restrict__ rank_loss,   // [B*P]
               float* __restrict__ posce_part)  // [B*TILES_B]
{
    extern __shared__ float sh[];               // ROWS_B*C_ data + 256 reduce
    const int tile = blockIdx.x;
    const int b    = blockIdx.y;
    const int tid  = threadIdx.x;
    const int row0 = tile * ROWS_B;
    const int rows = min(ROWS_B, P_ - row0);    // 128 or 28 (both mult of 4)

    // ---- async global -> LDS staging (ASYNCcnt path) ----
    const float* gsrc = conf_data + ((size_t)b * P_ + row0) * C_;  // 16B aligned
    unsigned ldsbase = (unsigned)(uintptr_t)(&sh[0]);
    const int nchunks = (rows * C_) >> 2;       // 16-byte chunks
    for (int i = tid; i < nchunks; i += 256) {
        const float* g = gsrc + (size_t)i * 4;
        unsigned     l = ldsbase + (unsigned)i * 16u;
        asm volatile("global_load_async_to_lds_b128 %0, %1, off"
                     :: "v"(l), "v"(g) : "memory");
    }
    // zero-fill rows beyond the tile so the WMMA feeds benign values
    for (int i = rows * C_ + tid; i < ROWS_B * C_; i += 256) sh[i] = 0.0f;
    asm volatile("s_wait_asynccnt 0" ::: "memory");
    __syncthreads();

    // ---- wave-cooperative LSE over 16 rows ----
    const int lane = tid & 31;
    const int wave = tid >> 5;
    const int half = lane >> 4;                 // 0: K even pair, 1: K odd pair
    const int mrow = lane & 15;                 // matrix row this lane feeds
    const float* xr = &sh[(wave * 16 + mrow) * C_];

    // per-row max (split halves, combine across lane^16)
    float mx = -INFINITY;
    {
        int kbeg = half * 40;
        for (int k = kbeg; k < kbeg + 40; ++k) mx = fmaxf(mx, xr[k]);
        if (half == 0) mx = fmaxf(mx, xr[80]);
        mx = fmaxf(mx, __shfl_xor(mx, 16, 32));
    }

    // Sum(exp) over k=0..79 via 20 chained WMMAs, B = ones
    v2f bones; bones.x = 1.0f; bones.y = 1.0f;
    v8f c = {};
#pragma unroll 4
    for (int kc = 0; kc < 20; ++kc) {
        int k0 = kc * 4 + 2 * half;
        v2f a;
        a.x = expf(xr[k0]     - mx);
        a.y = expf(xr[k0 + 1] - mx);
        c = __builtin_amdgcn_wmma_f32_16x16x4_f32(
                false, a, false, bones, (short)0, c, false, false);
    }

    // D layout: lanes 0-15 hold rows 0-7 in c[0..7]; lanes 16-31 hold rows 8-15.
    // Output lanes: 0-7 -> rows 0-7, 16-23 -> rows 8-15.
    float rs = 0.0f;
    {
        int jj = lane & 7;
#pragma unroll
        for (int j = 0; j < 8; ++j) if (jj == j) rs = c[j];
    }
    const bool isOut = (lane & 8) == 0;
    const int  orow  = ((lane >> 4) << 3) | (lane & 7);   // 0..15 within wave
    // fetch the owning row's max (lane that scanned it): src = lane + ((lane&16)>>1)
    float mout = __shfl(mx, lane + ((lane & 16) >> 1), 32);

    float pce = 0.0f;
    int rowInTile = wave * 16 + orow;
    if (isOut && rowInTile < rows) {
        const float* xo = &sh[rowInTile * C_];
        float ssum = rs + expf(xo[80] - mout);            // element 80 scalar
        float lse  = mout + logf(ssum);
        int   gidx = row0 + rowInTile;
        int   ct   = conf_t[(size_t)b * P_ + gidx];
        float ce   = lse - xo[ct];
        bool  pos  = ct > 0;
        rank_loss[(size_t)b * P_ + gidx] = pos ? 0.0f : ce;
        pce = pos ? ce : 0.0f;
    }

    // deterministic block reduction of positive-CE
    float* red = sh + ROWS_B * C_;
    red[tid] = pce;
    __syncthreads();
    for (int s = 128; s > 0; s >>= 1) {
        if (tid < s) red[tid] += red[tid + s];
        __syncthreads();
    }
    if (tid == 0) posce_part[b * TILES_B + tile] = red[0];
}

// ---------------------------------------------------------------------------
// Kernel C: hard-negative mining. In-LDS bitonic sort (ascending) of the
// per-batch rank_loss padded to 16384, then sum the K largest (ties don't
// change the sum). Fully deterministic.
// ---------------------------------------------------------------------------
__global__ __launch_bounds__(1024)
void mine_kernel(const float* __restrict__ rank_loss,
                 const int*   __restrict__ num_pos,
                 float* __restrict__ neg_part)   // [B]
{
    extern __shared__ float s[];                 // SORT_N + 1024
    const int b   = blockIdx.x;
    const int tid = threadIdx.x;

    for (int i = tid; i < SORT_N; i += 1024)
        s[i] = (i < P_) ? rank_loss[(size_t)b * P_ + i] : -1.0f;
    __syncthreads();

    for (int k = 2; k <= SORT_N; k <<= 1) {
        for (int j = k >> 1; j > 0; j >>= 1) {
            for (int i = tid; i < SORT_N; i += 1024) {
                int ixj = i ^ j;
                if (ixj > i) {
                    float a = s[i], c = s[ixj];
                    bool asc = ((i & k) == 0);
                    if (asc ? (a > c) : (a < c)) { s[i] = c; s[ixj] = a; }
                }
            }
            __syncthreads();
        }
    }

    int np = num_pos[b];
    int K  = min(3 * np, P_ - 1);
    float acc = 0.0f;
    for (int i = SORT_N - K + tid; i < SORT_N; i += 1024) acc += s[i];

    float* red = s + SORT_N;
    red[tid] = acc;
    __syncthreads();
    for (int t = 512; t > 0; t >>= 1) {
        if (tid < t) red[tid] += red[tid + t];
        __syncthreads();
    }
    if (tid == 0) neg_part[b] = red[0];
}

// ---------------------------------------------------------------------------
// Kernel D: final combine (single thread -> fully deterministic, trivial work)
// ---------------------------------------------------------------------------
__global__ void finish_kernel(const int*   __restrict__ num_pos,
                              const float* __restrict__ lossl_part,
                              const float* __restrict__ posce_part,
                              const float* __restrict__ neg_part,
                              float* __restrict__ out)
{
    if (threadIdx.x != 0 || blockIdx.x != 0) return;
    int N = 0; float ll = 0.0f, lc = 0.0f;
    for (int b = 0; b < B_; ++b) {
        N  += num_pos[b];
        ll += lossl_part[b];
        lc += neg_part[b];
    }
    for (int i = 0; i < B_ * TILES_B; ++i) lc += posce_part[i];
    float fN = (float)N;
    out[0] = ll / fN;
    out[1] = lc / fN;
}

// ---------------------------------------------------------------------------
extern "C" void kernel_launch(void* const* d_in, const int* in_sizes, int n_in,
                              void* d_out, int out_size, void* d_ws, size_t ws_size,
                              hipStream_t stream)
{
    const float* loc     = (const float*)d_in[0];
    const float* conf    = (const float*)d_in[1];
    const float* priors  = (const float*)d_in[2];
    const float* targets = (const float*)d_in[3];
    float* out = (float*)d_out;

    char* ws = (char*)d_ws;
    int*   w_conf_t = (int*)ws;    ws += (size_t)B_ * P_ * sizeof(int);
    float* w_rank   = (float*)ws;  ws += (size_t)B_ * P_ * sizeof(float);
    int*   w_npos   = (int*)ws;    ws += 256;
    float* w_lossl  = (float*)ws;  ws += 256;
    float* w_neg    = (float*)ws;  ws += 256;
    float* w_posce  = (float*)ws;  // B_*TILES_B floats

    size_t ldsA = (size_t)P_ * 8 + 64 * 4 + 16 * 4 + 16 * 4 + 16 * 8 + 256 * 4 + 256 * 4;
    size_t ldsB = (size_t)(ROWS_B * C_ + 256) * sizeof(float);
    size_t ldsC = (size_t)(SORT_N + 1024) * sizeof(float);

    match_kernel<<<dim3(B_), dim3(256), ldsA, stream>>>(
        loc, priors, targets, w_conf_t, w_npos, w_lossl);
    ce_kernel<<<dim3(TILES_B, B_), dim3(256), ldsB, stream>>>(
        conf, w_conf_t, w_rank, w_posce);
    mine_kernel<<<dim3(B_), dim3(1024), ldsC, stream>>>(
        w_rank, w_npos, w_neg);
    finish_kernel<<<dim3(1), dim3(32), 0, stream>>>(
        w_npos, w_lossl, w_posce, w_neg, out);
}